// ContrastiveLoss_29575144800392
// MI455X (gfx1250) — compile-verified
//
#include <hip/hip_runtime.h>
#include <math.h>

// Problem constants (match reference)
#define B_ROWS 4096
#define D_DIM  256
#define N_ROWS 8192      // 2B
#define TILE   128       // workgroup output tile (TILE x TILE)
#define KC     32        // K-panel staged in LDS per iteration
#define NP     (D_DIM / KC)  // 8 panels
#define LDA    40        // padded LDS row stride in floats (160B: 16B aligned, conflict-free)
#define MBLK   (N_ROWS / TILE)            // 64 tile-blocks per dimension
#define NTRI   ((MBLK * (MBLK + 1)) / 2)  // 2080 upper-triangular blocks

typedef __attribute__((ext_vector_type(2))) float v2f;
typedef __attribute__((ext_vector_type(8))) float v8f;
typedef __attribute__((ext_vector_type(4))) int   v4i;

#if __has_builtin(__builtin_amdgcn_global_load_async_to_lds_b128) && \
    __has_builtin(__builtin_amdgcn_s_wait_asynccnt)
#define USE_ASYNC_LDS 1
typedef __attribute__((address_space(1))) v4i* gas_v4i_ptr;
typedef __attribute__((address_space(3))) v4i* las_v4i_ptr;
#else
#define USE_ASYNC_LDS 0
#endif

// Monotone float -> uint encoding so unsigned atomicMax == float max.
__device__ __forceinline__ unsigned ord_enc(float f) {
  unsigned u = __float_as_uint(f);
  return (u & 0x80000000u) ? ~u : (u | 0x80000000u);
}
__device__ __forceinline__ float ord_dec(unsigned u) {
  return __uint_as_float((u & 0x80000000u) ? (u ^ 0x80000000u) : ~u);
}

// ---------------------------------------------------------------- init
__global__ void k_init(unsigned* __restrict__ negbits) {
  int i = blockIdx.x * blockDim.x + threadIdx.x;
  if (i < N_ROWS) negbits[i] = 0u;  // 0 encodes below every real float
}

// ------------------------------------------------------------ normalize
__global__ __launch_bounds__(256)
void k_normalize(const float* __restrict__ zi, const float* __restrict__ zj,
                 float* __restrict__ Nmat) {
  const int wave = threadIdx.x >> 5;
  const int lane = threadIdx.x & 31;
  const int row  = blockIdx.x * 8 + wave;
  const float* src = (row < B_ROWS) ? (zi + (size_t)row * D_DIM)
                                    : (zj + (size_t)(row - B_ROWS) * D_DIM);
  const float4* s4 = (const float4*)src;
  float4 a = s4[lane];
  float4 b = s4[lane + 32];
  float ss = a.x*a.x + a.y*a.y + a.z*a.z + a.w*a.w
           + b.x*b.x + b.y*b.y + b.z*b.z + b.w*b.w;
#pragma unroll
  for (int off = 16; off > 0; off >>= 1) ss += __shfl_xor(ss, off, 32);
  const float scl = 1.0f / fmaxf(sqrtf(ss), 1e-12f);
  a.x *= scl; a.y *= scl; a.z *= scl; a.w *= scl;
  b.x *= scl; b.y *= scl; b.z *= scl; b.w *= scl;
  float4* d4 = (float4*)(Nmat + (size_t)row * D_DIM);
  d4[lane]      = a;
  d4[lane + 32] = b;
}

// ------------------------------------------------------------ positives
__global__ __launch_bounds__(256)
void k_positives(const float* __restrict__ Nmat, float* __restrict__ pos) {
  const int wave = threadIdx.x >> 5;
  const int lane = threadIdx.x & 31;
  const int i = blockIdx.x * 8 + wave;
  const float4* p = (const float4*)(Nmat + (size_t)i * D_DIM);
  const float4* q = (const float4*)(Nmat + (size_t)(i + B_ROWS) * D_DIM);
  float4 a0 = p[lane], a1 = p[lane + 32];
  float4 b0 = q[lane], b1 = q[lane + 32];
  float d = a0.x*b0.x + a0.y*b0.y + a0.z*b0.z + a0.w*b0.w
          + a1.x*b1.x + a1.y*b1.y + a1.z*b1.z + a1.w*b1.w;
#pragma unroll
  for (int off = 16; off > 0; off >>= 1) d += __shfl_xor(d, off, 32);
  if (lane == 0) {
    const float v = d * 10.0f;   // / TEMPERATURE
    pos[i] = v;
    pos[i + B_ROWS] = v;
  }
}

// panel copy: 16 floats per thread (4 x b128), async-to-LDS when available
__device__ __forceinline__ void panel_copy(float* sdst, const float* gsrc) {
#if USE_ASYNC_LDS
#pragma unroll
  for (int i = 0; i < 4; ++i)
    __builtin_amdgcn_global_load_async_to_lds_b128(
        (gas_v4i_ptr)(gsrc + i * 4),
        (las_v4i_ptr)(sdst + i * 4), 0, 0);
#else
#pragma unroll
  for (int i = 0; i < 4; ++i)
    *(float4*)(sdst + i * 4) = *(const float4*)(gsrc + i * 4);
#endif
}
__device__ __forceinline__ void panel_commit() {
#if USE_ASYNC_LDS
  __builtin_amdgcn_s_wait_asynccnt(0);
#endif
  __syncthreads();
}

// ------------------------------------------------- WMMA sim-max GEMM
// Upper-triangular tiles only (sim symmetric): each tile feeds row-maxes AND
// column-maxes. 256 threads = 8 waves (4x2), wave patch 32x64 = 2x4 f32 WMMAs.
// Double-buffered LDS panels, prefetched via async global->LDS.
__global__ __launch_bounds__(256)
void k_simmax(const float* __restrict__ Nmat, unsigned* __restrict__ negbits) {
  __shared__ float As[2][TILE * LDA];   // 2 x 20 KB
  __shared__ float Bs[2][TILE * LDA];   // 2 x 20 KB

  // map linear block id -> (by, bx) with by <= bx
  const int lin = blockIdx.x;
  int by = (int)(((float)(2 * MBLK + 1) -
                  sqrtf((float)((2 * MBLK + 1) * (2 * MBLK + 1) - 8 * lin))) * 0.5f);
  by = by < 0 ? 0 : (by >= MBLK ? MBLK - 1 : by);
  while ((by + 1) * MBLK - ((by + 1) * by) / 2 <= lin) ++by;
  while (by * MBLK - (by * (by - 1)) / 2 > lin) --by;
  const int bx = by + (lin - (by * MBLK - (by * (by - 1)) / 2));
  const int rowBase = by * TILE;
  const int colBase = bx * TILE;

  const int t    = threadIdx.x;
  const int wave = t >> 5;
  const int lane = t & 31;
  const int waveM = (wave >> 1) * 32;   // 0,32,64,96
  const int waveN = (wave & 1)  * 64;   // 0,64
  const int l16   = lane & 15;
  const int koff  = (lane >> 4) << 1;   // 16x4 frag: lanes 0-15 -> K0,1 ; 16-31 -> K2,3

  const v8f zero = {0.f,0.f,0.f,0.f,0.f,0.f,0.f,0.f};
  v8f acc[2][4];
#pragma unroll
  for (int mi = 0; mi < 2; ++mi)
#pragma unroll
    for (int ni = 0; ni < 4; ++ni) acc[mi][ni] = zero;

  // tile loaders: 2 threads per row, 16 floats each
  const int lr = t >> 1;            // 0..127
  const int lc = (t & 1) * 16;      // 0 or 16
  const float* gA = Nmat + (size_t)(rowBase + lr) * D_DIM + lc;
  const float* gB = Nmat + (size_t)(colBase + lr) * D_DIM + lc;
  const int soff = lr * LDA + lc;

  // prologue: panel 0 into buffer 0
  panel_copy(&As[0][soff], gA);
  panel_copy(&Bs[0][soff], gB);
  panel_commit();

  for (int p = 0; p < NP; ++p) {
    const int cur = p & 1, nxt = cur ^ 1;
#if !USE_ASYNC_LDS
    float4 rA[4], rB[4];
#endif
    if (p + 1 < NP) {
#if USE_ASYNC_LDS
      panel_copy(&As[nxt][soff], gA + (p + 1) * KC);
      panel_copy(&Bs[nxt][soff], gB + (p + 1) * KC);
#else
#pragma unroll
      for (int i = 0; i < 4; ++i) {
        rA[i] = *(const float4*)(gA + (p + 1) * KC + i * 4);
        rB[i] = *(const float4*)(gB + (p + 1) * KC + i * 4);
      }
#endif
    }

    const float* Ac = &As[cur][0];
    const float* Bc = &Bs[cur][0];
#pragma unroll
    for (int kk = 0; kk < KC; kk += 4) {
      v2f afr[2], bfr[4];
#pragma unroll
      for (int mi = 0; mi < 2; ++mi)
        afr[mi] = *(const v2f*)&Ac[(waveM + mi * 16 + l16) * LDA + kk + koff];
#pragma unroll
      for (int ni = 0; ni < 4; ++ni)
        bfr[ni] = *(const v2f*)&Bc[(waveN + ni * 16 + l16) * LDA + kk + koff];
#pragma unroll
      for (int mi = 0; mi < 2; ++mi)
#pragma unroll
        for (int ni = 0; ni < 4; ++ni)
          acc[mi][ni] = __builtin_amdgcn_wmma_f32_16x16x4_f32(
              false, afr[mi], false, bfr[ni], (short)0, acc[mi][ni], false, false);
    }

#if !USE_ASYNC_LDS
    if (p + 1 < NP) {
#pragma unroll
      for (int i = 0; i < 4; ++i) {
        *(float4*)(&As[nxt][soff] + i * 4) = rA[i];
        *(float4*)(&Bs[nxt][soff] + i * 4) = rB[i];
      }
    }
#endif
    panel_commit();
  }

  // Epilogue: scale by 1/T, mask diagonal, per-row AND per-column maxes.
  // C layout (16x16 f32): VGPR v, lanes 0-15 -> M=v, N=lane; lanes 16-31 -> M=v+8, same N.
  float colmax[4];
#pragma unroll
  for (int ni = 0; ni < 4; ++ni) colmax[ni] = -__builtin_inff();

#pragma unroll
  for (int mi = 0; mi < 2; ++mi) {
    const int rbase = rowBase + waveM + mi * 16 + ((lane >> 4) << 3);
#pragma unroll
    for (int v = 0; v < 8; ++v) {
      const int grow = rbase + v;
      float rowm = -__builtin_inff();
#pragma unroll
      for (int ni = 0; ni < 4; ++ni) {
        const int gcol = colBase + waveN + ni * 16 + l16;
        float s = acc[mi][ni][v] * 10.0f;
        s = (grow == gcol) ? -__builtin_inff() : s;
        rowm = fmaxf(rowm, s);
        colmax[ni] = fmaxf(colmax[ni], s);
      }
#pragma unroll
      for (int off = 1; off < 16; off <<= 1)
        rowm = fmaxf(rowm, __shfl_xor(rowm, off, 32));
      if (l16 == 0) atomicMax(&negbits[grow], ord_enc(rowm));
    }
  }
  // transposed contribution (sim symmetric): column max -> negbits[col]
#pragma unroll
  for (int ni = 0; ni < 4; ++ni) {
    const float cm = fmaxf(colmax[ni], __shfl_xor(colmax[ni], 16, 32));
    if (lane < 16) atomicMax(&negbits[colBase + waveN + ni * 16 + l16], ord_enc(cm));
  }
}

// ------------------------------------------------------------- finalize
__global__ __launch_bounds__(1024)
void k_finalize(const unsigned* __restrict__ negbits,
                const float* __restrict__ pos, float* __restrict__ out) {
  __shared__ float sm[1024];
  __shared__ float s1[1024];
  __shared__ float s2[1024];
  const int t = threadIdx.x;
  float pm = -__builtin_inff();
  float nsum = 0.0f;
  for (int i = t; i < N_ROWS; i += 1024) {
    pm = fmaxf(pm, pos[i]);
    nsum += ord_dec(negbits[i]);
  }
  sm[t] = pm;
  __syncthreads();
  for (int s = 512; s > 0; s >>= 1) {
    if (t < s) sm[t] = fmaxf(sm[t], sm[t + s]);
    __syncthreads();
  }
  const float mx = sm[0];
  float es = 0.0f;
  for (int i = t; i < N_ROWS; i += 1024) es += expf(pos[i] - mx);
  s1[t] = es;
  s2[t] = nsum;
  __syncthreads();
  for (int s = 512; s > 0; s >>= 1) {
    if (t < s) { s1[t] += s1[t + s]; s2[t] += s2[t + s]; }
    __syncthreads();
  }
  if (t == 0) out[0] = s2[0] / (float)N_ROWS - (mx + logf(s1[0]));
}

// -------------------------------------------------------------- launch
extern "C" void kernel_launch(void* const* d_in, const int* in_sizes, int n_in,
                              void* d_out, int out_size, void* d_ws, size_t ws_size,
                              hipStream_t stream) {
  (void)in_sizes; (void)n_in; (void)out_size; (void)ws_size;
  const float* zi = (const float*)d_in[0];
  const float* zj = (const float*)d_in[1];
  float* out = (float*)d_out;

  // workspace layout: N (8 MB) | positives (32 KB) | negmax bits (32 KB)
  float*    Nmat    = (float*)d_ws;
  float*    posArr  = (float*)((char*)d_ws + (size_t)N_ROWS * D_DIM * sizeof(float));
  unsigned* negbits = (unsigned*)((char*)posArr + (size_t)N_ROWS * sizeof(float));

  k_init<<<(N_ROWS + 255) / 256, 256, 0, stream>>>(negbits);
  k_normalize<<<N_ROWS / 8, 256, 0, stream>>>(zi, zj, Nmat);
  k_positives<<<B_ROWS / 8, 256, 0, stream>>>(Nmat, posArr);
  k_simmax<<<NTRI, 256, 0, stream>>>(Nmat, negbits);   // 2080 upper-tri tiles
  k_finalize<<<1, 1024, 0, stream>>>(negbits, posArr, out);
}